// QuantumTransformerBlock_65481071407659
// MI455X (gfx1250) — compile-verified
//
#include <hip/hip_runtime.h>
#include <math.h>

typedef float v2f __attribute__((ext_vector_type(2)));
typedef float v8f __attribute__((ext_vector_type(8)));

#define LN_EPS 1e-5f

__device__ __forceinline__ float wave_reduce_add(float v) {
#pragma unroll
  for (int m = 16; m >= 1; m >>= 1) v += __shfl_xor(v, m, 32);
  return v;
}

// ---------------------------------------------------------------------------
// Kernel 1: per-token 8-qubit statevector simulation + residual + LayerNorm1.
// One token per wave32. Amp index a = (lane[4:0] << 3) | j[2:0].
// Wires 0..2 -> local bits j, wires 3..7 -> lane bits.
// ---------------------------------------------------------------------------
__global__ __launch_bounds__(256)
void quantum_ln1_kernel(const float* __restrict__ x,
                        const float* __restrict__ wts,
                        const float* __restrict__ gamma1,
                        const float* __restrict__ beta1,
                        float* __restrict__ x1,
                        int T)
{
  const int lane = threadIdx.x & 31;
  const int wv   = threadIdx.x >> 5;
  const int t    = blockIdx.x * 8 + wv;
  if (t >= T) return;

  // token inputs (reused for RX angles and the LN1 residual)
  float xv[8];
#pragma unroll
  for (int w = 0; w < 8; ++w) xv[w] = x[t * 8 + w];

  // RX(x) embedding factors: f(0) = c, f(1) = -i*s  (th = x/2)
  // args are O(1) -> native hardware sin/cos is plenty accurate here
  float cx[8], sx[8];
#pragma unroll
  for (int w = 0; w < 8; ++w) {
    float s, c;
    __sincosf(0.5f * xv[w], &s, &c);
    sx[w] = s; cx[w] = c;
  }

  // Local (wires 0..2) products for the 8 in-lane amplitudes
  float Lre[8], Lim[8];
#pragma unroll
  for (int j = 0; j < 8; ++j) {
    float pr = 1.f, pi = 0.f;
#pragma unroll
    for (int w = 0; w < 3; ++w) {
      float nr, ni;
      if ((j >> w) & 1) { nr = sx[w] * pi; ni = -sx[w] * pr; }  // *(-i s)
      else              { nr = cx[w] * pr; ni =  cx[w] * pi; }  // *c
      pr = nr; pi = ni;
    }
    Lre[j] = pr; Lim[j] = pi;
  }
  // Lane prefactor (wires 3..7)
  float Pr = 1.f, Pi = 0.f;
#pragma unroll
  for (int w = 3; w < 8; ++w) {
    const bool b = (lane >> (w - 3)) & 1;
    float nr = b ?  sx[w] * Pi : cx[w] * Pr;
    float ni = b ? -sx[w] * Pr : cx[w] * Pi;
    Pr = nr; Pi = ni;
  }
  float re[8], im[8];
#pragma unroll
  for (int j = 0; j < 8; ++j) {
    re[j] = Pr * Lre[j] - Pi * Lim[j];
    im[j] = Pr * Lim[j] + Pi * Lre[j];
  }

  // Two entangling layers: RY(weights) per wire, then CNOT ring
#pragma unroll
  for (int l = 0; l < 2; ++l) {
    // RY on wires 0..2 (in-lane pairs)
#pragma unroll
    for (int w = 0; w < 3; ++w) {
      float s, c;
      __sincosf(0.5f * wts[l * 8 + w], &s, &c);
      const int m = 1 << w;
#pragma unroll
      for (int j = 0; j < 8; ++j) {
        if (!(j & m)) {
          const int k = j | m;
          float r0 = re[j], i0 = im[j], r1 = re[k], i1 = im[k];
          re[j] = c * r0 - s * r1; im[j] = c * i0 - s * i1;
          re[k] = s * r0 + c * r1; im[k] = s * i0 + c * i1;
        }
      }
    }
    // RY on wires 3..7 (cross-lane)
#pragma unroll
    for (int w = 3; w < 8; ++w) {
      float s, c;
      __sincosf(0.5f * wts[l * 8 + w], &s, &c);
      const int lm = 1 << (w - 3);
      const float sg = (lane & lm) ? s : -s;
#pragma unroll
      for (int j = 0; j < 8; ++j) {
        float orr = __shfl_xor(re[j], lm, 32);
        float oii = __shfl_xor(im[j], lm, 32);
        re[j] = c * re[j] + sg * orr;
        im[j] = c * im[j] + sg * oii;
      }
    }
    // CNOT(0,1): bit0==1 -> flip bit1 : swap (1,3),(5,7)
    { float tt;
      tt=re[1]; re[1]=re[3]; re[3]=tt;  tt=im[1]; im[1]=im[3]; im[3]=tt;
      tt=re[5]; re[5]=re[7]; re[7]=tt;  tt=im[5]; im[5]=im[7]; im[7]=tt; }
    // CNOT(1,2): bit1==1 -> flip bit2 : swap (2,6),(3,7)
    { float tt;
      tt=re[2]; re[2]=re[6]; re[6]=tt;  tt=im[2]; im[2]=im[6]; im[6]=tt;
      tt=re[3]; re[3]=re[7]; re[7]=tt;  tt=im[3]; im[3]=im[7]; im[7]=tt; }
    // CNOT(2,3): local bit2==1 -> swap with lane^1 (unconditional for j>=4)
#pragma unroll
    for (int j = 4; j < 8; ++j) {
      re[j] = __shfl_xor(re[j], 1, 32);
      im[j] = __shfl_xor(im[j], 1, 32);
    }
    // CNOT(3,4),(4,5),(5,6),(6,7): lane-bit control -> lane swap if control set
#pragma unroll
    for (int c2 = 3; c2 < 7; ++c2) {
      const int cm = 1 << (c2 - 3);
      const int tm = 2 << (c2 - 3);
      const bool ctl = (lane & cm);
#pragma unroll
      for (int j = 0; j < 8; ++j) {
        float orr = __shfl_xor(re[j], tm, 32);
        float oii = __shfl_xor(im[j], tm, 32);
        re[j] = ctl ? orr : re[j];
        im[j] = ctl ? oii : im[j];
      }
    }
    // CNOT(7,0): lane bit4 set -> swap local pairs (j, j^1)
    { const bool ctl = (lane & 16);
#pragma unroll
      for (int j = 0; j < 8; j += 2) {
        float a = re[j], b = re[j + 1];
        re[j]     = ctl ? b : a;
        re[j + 1] = ctl ? a : b;
        float ai = im[j], bi = im[j + 1];
        im[j]     = ctl ? bi : ai;
        im[j + 1] = ctl ? ai : bi;
      }
    }
  }

  // probabilities and <Z_w>
  float p[8];
#pragma unroll
  for (int j = 0; j < 8; ++j) p[j] = re[j] * re[j] + im[j] * im[j];

  float ez[8];
#pragma unroll
  for (int w = 0; w < 3; ++w) {
    float v = 0.f;
#pragma unroll
    for (int j = 0; j < 8; ++j) v += ((j >> w) & 1) ? -p[j] : p[j];
    ez[w] = wave_reduce_add(v);
  }
  float ps = 0.f;
#pragma unroll
  for (int j = 0; j < 8; ++j) ps += p[j];
#pragma unroll
  for (int w = 3; w < 8; ++w) {
    float v = ((lane >> (w - 3)) & 1) ? -ps : ps;
    ez[w] = wave_reduce_add(v);
  }

  // residual + LayerNorm1 (values are wave-uniform)
  float v8a[8];
  float mean = 0.f;
#pragma unroll
  for (int w = 0; w < 8; ++w) { v8a[w] = xv[w] + ez[w]; mean += v8a[w]; }
  mean *= 0.125f;
  float var = 0.f;
#pragma unroll
  for (int w = 0; w < 8; ++w) { float d = v8a[w] - mean; var += d * d; }
  var *= 0.125f;
  const float inv = rsqrtf(var + LN_EPS);
  float y[8];
#pragma unroll
  for (int w = 0; w < 8; ++w) y[w] = (v8a[w] - mean) * inv * gamma1[w] + beta1[w];

  float ov = y[0];
#pragma unroll
  for (int w = 1; w < 8; ++w) ov = (lane == w) ? y[w] : ov;
  if (lane < 8) x1[t * 8 + lane] = ov;
}

// ---------------------------------------------------------------------------
// Kernel 2: WMMA FFN (x1@W1 -> relu -> @W2) + residual + LayerNorm2.
// 16 tokens per wave, V_WMMA_F32_16X16X4_F32. 8 waves / 128 tokens per block.
// ---------------------------------------------------------------------------
__global__ __launch_bounds__(256)
void ffn_ln2_kernel(const float* __restrict__ x1g,
                    const float* __restrict__ W1,
                    const float* __restrict__ b1,
                    const float* __restrict__ W2,
                    const float* __restrict__ b2,
                    const float* __restrict__ gamma2,
                    const float* __restrict__ beta2,
                    float* __restrict__ out,
                    int T)
{
  __shared__ __align__(16) float x1s[128 * 8];
  __shared__ __align__(16) float Hs[8][16][36];   // pad 36 -> conflict-free

  const int lane = threadIdx.x & 31;
  const int wv   = threadIdx.x >> 5;
  const int blockBase = blockIdx.x * 128;
  const int nTok = min(128, T - blockBase);

  // stage x1 tile into LDS (also makes the x1==out fallback race-free)
  for (int i = threadIdx.x; i < 128 * 8; i += 256)
    x1s[i] = (i < nTok * 8) ? x1g[blockBase * 8 + i] : 0.f;
  __syncthreads();

  const int n  = lane & 15;       // N / column index
  const int kh = lane >> 4;       // K half-select
  const int rowTok = wv * 16 + n; // A-row token (local)

  // A (16x8 of x1) split into two 16x4 K-chunks; ISA f32 A layout:
  // lanes 0-15: K = 0,1 ; lanes 16-31: K = 2,3
  v2f a0 = *(const v2f*)&x1s[rowTok * 8 + 2 * kh];
  v2f a1 = *(const v2f*)&x1s[rowTok * 8 + 4 + 2 * kh];

  // ---- matmul1: H = relu(x1 @ W1 + b1), N tiled as 2 x 16 ----
  float hval[2][8];
#pragma unroll
  for (int nt = 0; nt < 2; ++nt) {
    const int f = nt * 16 + n;
    v2f bA, bB;
    bA.x = W1[(2 * kh)     * 32 + f];
    bA.y = W1[(2 * kh + 1) * 32 + f];
    bB.x = W1[(4 + 2 * kh)     * 32 + f];
    bB.y = W1[(4 + 2 * kh + 1) * 32 + f];
    v8f acc = {0.f, 0.f, 0.f, 0.f, 0.f, 0.f, 0.f, 0.f};
    acc = __builtin_amdgcn_wmma_f32_16x16x4_f32(false, a0, false, bA,
                                                (short)0, acc, false, false);
    acc = __builtin_amdgcn_wmma_f32_16x16x4_f32(false, a1, false, bB,
                                                (short)0, acc, false, false);
    const float bb = b1[f];
#pragma unroll
    for (int r = 0; r < 8; ++r) {
      float h = acc[r] + bb;
      hval[nt][r] = h > 0.f ? h : 0.f;
    }
  }

  // D-layout (vgpr r: rows r / r+8) -> LDS A-layout staging
  const int mrow = kh << 3;
#pragma unroll
  for (int nt = 0; nt < 2; ++nt)
#pragma unroll
    for (int r = 0; r < 8; ++r)
      Hs[wv][r + mrow][nt * 16 + n] = hval[nt][r];
  __syncthreads();

  // ---- matmul2: Y = H(16x32) @ W2(32x8), N zero-padded to 16 ----
  const bool nval = n < 8;
  v8f acc2 = {0.f, 0.f, 0.f, 0.f, 0.f, 0.f, 0.f, 0.f};
#pragma unroll
  for (int kk = 0; kk < 8; ++kk) {
    const int k0 = 4 * kk + 2 * kh;
    v2f aa = *(const v2f*)&Hs[wv][n][k0];
    v2f bb;
    bb.x = nval ? W2[k0 * 8 + n]       : 0.f;
    bb.y = nval ? W2[(k0 + 1) * 8 + n] : 0.f;
    acc2 = __builtin_amdgcn_wmma_f32_16x16x4_f32(false, aa, false, bb,
                                                 (short)0, acc2, false, false);
  }

  // ---- bias + residual + LayerNorm2 per token row (8 valid lanes/row) ----
  const float g2v = nval ? gamma2[n] : 0.f;
  const float be2 = nval ? beta2[n]  : 0.f;
  const float b2v = nval ? b2[n]     : 0.f;
#pragma unroll
  for (int r = 0; r < 8; ++r) {
    const int m   = r + mrow;
    const int tok = blockBase + wv * 16 + m;
    float val = acc2[r] + b2v + (nval ? x1s[(wv * 16 + m) * 8 + n] : 0.f);
    val = nval ? val : 0.f;
    float s = val;
    s += __shfl_xor(s, 1, 32); s += __shfl_xor(s, 2, 32); s += __shfl_xor(s, 4, 32);
    float q = val * val;
    q += __shfl_xor(q, 1, 32); q += __shfl_xor(q, 2, 32); q += __shfl_xor(q, 4, 32);
    const float mean = s * 0.125f;
    const float var  = q * 0.125f - mean * mean;
    const float inv  = rsqrtf(var + LN_EPS);
    const float res  = (val - mean) * inv * g2v + be2;
    if (nval && tok < T) out[tok * 8 + n] = res;
  }
}

extern "C" void kernel_launch(void* const* d_in, const int* in_sizes, int n_in,
                              void* d_out, int out_size, void* d_ws, size_t ws_size,
                              hipStream_t stream) {
  const float* x      = (const float*)d_in[0];
  const float* wts    = (const float*)d_in[1];
  const float* gamma1 = (const float*)d_in[2];
  const float* beta1  = (const float*)d_in[3];
  const float* gamma2 = (const float*)d_in[4];
  const float* beta2  = (const float*)d_in[5];
  const float* W1     = (const float*)d_in[6];
  const float* b1     = (const float*)d_in[7];
  const float* W2     = (const float*)d_in[8];
  const float* b2     = (const float*)d_in[9];
  float* out = (float*)d_out;

  const int T = in_sizes[0] / 8;   // B*S tokens
  float* x1 = (ws_size >= (size_t)T * 8 * sizeof(float)) ? (float*)d_ws : out;

  const int g1 = (T + 7) / 8;      // 8 waves (tokens) per block
  quantum_ln1_kernel<<<g1, 256, 0, stream>>>(x, wts, gamma1, beta1, x1, T);

  const int g2 = (T + 127) / 128;  // 128 tokens per block
  ffn_ln2_kernel<<<g2, 256, 0, stream>>>(x1, W1, b1, W2, b2,
                                         gamma2, beta2, out, T);
}